// MeshReduce_42528766165303
// MI455X (gfx1250) — compile-verified
//
#include <hip/hip_runtime.h>
#include <hip/hip_bf16.h>

typedef __attribute__((ext_vector_type(2))) float v2f;
typedef __attribute__((ext_vector_type(8))) float v8f;

#define EPS_LN 1e-5f
#define EPS_W  1e-16f
#define FLT_BIG 3.4e38f

// Branchless sorted top-3 insert: 3 compares + 10 selects, EXEC untouched.
__device__ __forceinline__ void top3_insert(float s, int j,
                                            float& d0, float& d1, float& d2,
                                            int& j0, int& j1, int& j2) {
    const bool l0 = s < d0, l1 = s < d1, l2 = s < d2;
    d2 = l1 ? d1 : (l2 ? s : d2);
    j2 = l1 ? j1 : (l2 ? j : j2);
    d1 = l0 ? d0 : (l1 ? s : d1);
    j1 = l0 ? j0 : (l1 ? j : j1);
    d0 = l0 ? s  : d0;
    j0 = l0 ? j  : j0;
}

__device__ __forceinline__ float wave_sum32(float v) {
    #pragma unroll
    for (int off = 16; off > 0; off >>= 1) v += __shfl_xor(v, off, 32);
    return v;
}

// ---------------------------------------------------------------------------
// kNN (k=3) partial pass via V_WMMA_F32_16X16X4_F32, candidate-sliced.
// One wave = 16 queries x one candidate slice (tps tiles of 16 candidates).
//   A (16x4, M=candidate):  row m = (cx, cy, cz, |c|^2)
//   B (4x16, N=query):      col n = (-2qx, -2qy, -2qz, 1)
//   => S[m][n] = |c_m|^2 - 2 c_m . q_n   (d^2 minus rank-invariant |q|^2)
// Per tile: WMMA -> per-lane min of 8 scores -> wave-uniform __any() gate ->
// (rare) 8 branchless inserts. Half-waves merged via shfl_xor(16); lanes 0-15
// write this slice's partial top-3 (score + candidate index) to workspace.
// ---------------------------------------------------------------------------
__global__ __launch_bounds__(256) void knn3_wmma_partial_kernel(
    const float* __restrict__ posQ, int nQ,
    const float* __restrict__ posC, int nC,
    int nSlices, int tps,
    float* __restrict__ partD, int* __restrict__ partJ) {

    const int wave  = (blockIdx.x * 256 + threadIdx.x) >> 5;
    const int lane  = threadIdx.x & 31;
    const int group = wave / nSlices;
    const int slice = wave - group * nSlices;
    const int qbase = group * 16;
    if (qbase >= nQ) return;       // wave-uniform exit; EXEC all-ones below

    const int   qi = qbase + (lane & 15);
    const int   h  = lane >> 4;    // 0: lanes 0-15 (K=0,1), 1: lanes 16-31 (K=2,3)
    const float qx = posQ[qi * 3 + 0];
    const float qy = posQ[qi * 3 + 1];
    const float qz = posQ[qi * 3 + 2];

    v2f bmat;
    bmat[0] = h ? (-2.0f * qz) : (-2.0f * qx);
    bmat[1] = h ? 1.0f         : (-2.0f * qy);

    float d0 = FLT_BIG, d1 = FLT_BIG, d2 = FLT_BIG;
    int   j0 = 0,       j1 = 0,       j2 = 0;

    const int tiles  = nC >> 4;                  // nC multiple of 16
    const int tStart = slice * tps;
    const int tEnd   = min(tiles, tStart + tps);

    for (int t = tStart; t < tEnd; ++t) {
        const int ci = (t << 4) + (lane & 15);
        __builtin_prefetch(posC + (ci + 16) * 3, 0, 3);   // global_prefetch_b8
        const float cx = posC[ci * 3 + 0];
        const float cy = posC[ci * 3 + 1];
        const float cz = posC[ci * 3 + 2];
        const float cn = cx * cx + cy * cy + cz * cz;

        v2f amat;
        amat[0] = h ? cz : cx;
        amat[1] = h ? cn : cy;

        v8f acc = {};
        acc = __builtin_amdgcn_wmma_f32_16x16x4_f32(
            /*neg_a=*/false, amat, /*neg_b=*/false, bmat,
            /*c_mod=*/(short)0, acc, /*reuse_a=*/false, /*reuse_b=*/false);

        // Fast reject: does any lane improve its top-3 this tile? (rare)
        const float m = fminf(fminf(fminf(acc[0], acc[1]), fminf(acc[2], acc[3])),
                              fminf(fminf(acc[4], acc[5]), fminf(acc[6], acc[7])));
        if (__any(m < d2)) {
            const int mbase = (t << 4) + (h << 3);
            #pragma unroll
            for (int r = 0; r < 8; ++r)
                top3_insert(acc[r], mbase + r, d0, d1, d2, j0, j1, j2);
        }
    }

    // Merge the two half-wave candidate sets (same query, disjoint M ranges).
    float e0 = __shfl_xor(d0, 16, 32); int f0 = __shfl_xor(j0, 16, 32);
    float e1 = __shfl_xor(d1, 16, 32); int f1 = __shfl_xor(j1, 16, 32);
    float e2 = __shfl_xor(d2, 16, 32); int f2 = __shfl_xor(j2, 16, 32);
    top3_insert(e0, f0, d0, d1, d2, j0, j1, j2);
    top3_insert(e1, f1, d0, d1, d2, j0, j1, j2);
    top3_insert(e2, f2, d0, d1, d2, j0, j1, j2);

    if (h == 0) {
        const int base = (qi * nSlices + slice) * 3;
        partD[base + 0] = d0; partJ[base + 0] = j0;
        partD[base + 1] = d1; partJ[base + 1] = j1;
        partD[base + 2] = d2; partJ[base + 2] = j2;
    }
}

// ---------------------------------------------------------------------------
// Merge per-slice partial top-3 lists and emit final idx + 1/max(d^2,eps).
// One thread per query; exact d^2 recomputed from the winning positions.
// ---------------------------------------------------------------------------
__global__ __launch_bounds__(256) void knn3_merge_kernel(
    const float* __restrict__ posQ, int nQ,
    const float* __restrict__ posC,
    const float* __restrict__ partD, const int* __restrict__ partJ,
    int nSlices,
    int* __restrict__ outIdx, float* __restrict__ outW) {

    const int q = blockIdx.x * 256 + threadIdx.x;
    if (q >= nQ) return;

    float d0 = FLT_BIG, d1 = FLT_BIG, d2 = FLT_BIG;
    int   j0 = 0,       j1 = 0,       j2 = 0;
    for (int s = 0; s < nSlices; ++s) {
        const int base = (q * nSlices + s) * 3;
        top3_insert(partD[base + 0], partJ[base + 0], d0, d1, d2, j0, j1, j2);
        top3_insert(partD[base + 1], partJ[base + 1], d0, d1, d2, j0, j1, j2);
        top3_insert(partD[base + 2], partJ[base + 2], d0, d1, d2, j0, j1, j2);
    }

    const float qx = posQ[q * 3 + 0];
    const float qy = posQ[q * 3 + 1];
    const float qz = posQ[q * 3 + 2];
    const int jj[3] = {j0, j1, j2};
    #pragma unroll
    for (int k = 0; k < 3; ++k) {
        const int j = jj[k];
        const float dx = posC[j * 3 + 0] - qx;
        const float dy = posC[j * 3 + 1] - qy;
        const float dz = posC[j * 3 + 2] - qz;
        const float dd = dx * dx + dy * dy + dz * dz;
        outIdx[q * 3 + k] = j;
        outW[q * 3 + k]   = 1.0f / fmaxf(dd, EPS_W);
    }
}

// ---------------------------------------------------------------------------
// Stage 2: piv[b,q,:] = sum_k w_k * LayerNorm(nf[b*Nm + idx[q,k], :]) / sum_k w_k
// One wave per (b,q); C == 128 -> one float4 per lane. LN computed only for
// gathered rows (full LN over all 160K rows never materialized).
// ---------------------------------------------------------------------------
__global__ __launch_bounds__(256) void piv_gather_ln_kernel(
    const float* __restrict__ nf, const float* __restrict__ gamma,
    const float* __restrict__ beta, const int* __restrict__ idx,
    const float* __restrict__ w, float* __restrict__ piv,
    int B, int Np, int Nm, int C) {

    const int wave = (blockIdx.x * 256 + threadIdx.x) >> 5;
    const int lane = threadIdx.x & 31;
    if (wave >= B * Np) return;
    const int b = wave / Np;
    const int q = wave - b * Np;

    const float w0 = w[q * 3 + 0], w1 = w[q * 3 + 1], w2 = w[q * 3 + 2];
    const float inv_wsum = 1.0f / (w0 + w1 + w2);
    const float wk[3] = {w0, w1, w2};

    const float4 g4 = ((const float4*)gamma)[lane];
    const float4 b4 = ((const float4*)beta)[lane];

    float4 acc = make_float4(0.f, 0.f, 0.f, 0.f);
    const float invC = 1.0f / (float)C;

    #pragma unroll
    for (int k = 0; k < 3; ++k) {
        const size_t row = (size_t)b * Nm + idx[q * 3 + k];
        const float4 x = ((const float4*)(nf + row * C))[lane];
        float s  = x.x + x.y + x.z + x.w;
        float s2 = x.x * x.x + x.y * x.y + x.z * x.z + x.w * x.w;
        s  = wave_sum32(s);
        s2 = wave_sum32(s2);
        const float mu   = s * invC;
        const float var  = fmaxf(s2 * invC - mu * mu, 0.0f);
        const float rstd = rsqrtf(var + EPS_LN);
        const float c    = wk[k];
        acc.x += c * ((x.x - mu) * rstd * g4.x + b4.x);
        acc.y += c * ((x.y - mu) * rstd * g4.y + b4.y);
        acc.z += c * ((x.z - mu) * rstd * g4.z + b4.z);
        acc.w += c * ((x.w - mu) * rstd * g4.w + b4.w);
    }
    acc.x *= inv_wsum; acc.y *= inv_wsum; acc.z *= inv_wsum; acc.w *= inv_wsum;
    ((float4*)(piv + ((size_t)(b * Np + q)) * C))[lane] = acc;
}

// ---------------------------------------------------------------------------
// Stage 3: out[b*Nm+n,:] = sum_k w2_k * piv[b, idx2[n,k], :] / sum_k w2_k
// One wave per (b,n); piv (4 MB) is L2-resident on a 192 MB L2.
// ---------------------------------------------------------------------------
__global__ __launch_bounds__(256) void mesh_gather_kernel(
    const float* __restrict__ piv, const int* __restrict__ idx,
    const float* __restrict__ w, float* __restrict__ out,
    int B, int Np, int Nm, int C) {

    const int wave = (blockIdx.x * 256 + threadIdx.x) >> 5;
    const int lane = threadIdx.x & 31;
    if (wave >= B * Nm) return;
    const int b = wave / Nm;
    const int n = wave - b * Nm;

    const float w0 = w[n * 3 + 0], w1 = w[n * 3 + 1], w2 = w[n * 3 + 2];
    const float inv_wsum = 1.0f / (w0 + w1 + w2);
    const float wk[3] = {w0, w1, w2};

    float4 acc = make_float4(0.f, 0.f, 0.f, 0.f);
    #pragma unroll
    for (int k = 0; k < 3; ++k) {
        const size_t row = (size_t)b * Np + idx[n * 3 + k];
        const float4 p = ((const float4*)(piv + row * C))[lane];
        const float c = wk[k];
        acc.x += c * p.x; acc.y += c * p.y; acc.z += c * p.z; acc.w += c * p.w;
    }
    acc.x *= inv_wsum; acc.y *= inv_wsum; acc.z *= inv_wsum; acc.w *= inv_wsum;
    ((float4*)(out + ((size_t)(b * Nm + n)) * C))[lane] = acc;
}

extern "C" void kernel_launch(void* const* d_in, const int* in_sizes, int n_in,
                              void* d_out, int out_size, void* d_ws, size_t ws_size,
                              hipStream_t stream) {
    const float* nf    = (const float*)d_in[0];  // [B*Nm, C]
    const float* gamma = (const float*)d_in[1];  // [C]
    const float* beta  = (const float*)d_in[2];  // [C]
    const float* posM  = (const float*)d_in[3];  // [Nm, 3]
    const float* posP  = (const float*)d_in[4];  // [Np, 3]

    const int C  = in_sizes[1];
    const int Nm = in_sizes[3] / 3;
    const int Np = in_sizes[4] / 3;
    const int B  = in_sizes[0] / (Nm * C);       // batch_size derived on host

    // Candidate slicing: ~128 tiles (2048 candidates) per wave-slice.
    const int tiles1 = Nm >> 4, tiles2 = Np >> 4;
    const int nSl1 = (tiles1 + 127) / 128;       // pass1: 2500 tiles -> 20 slices
    const int nSl2 = (tiles2 + 127) / 128;       // pass2: 125 tiles  -> 1 slice
    const int tps1 = (tiles1 + nSl1 - 1) / nSl1;
    const int tps2 = (tiles2 + nSl2 - 1) / nSl2;

    // Workspace carve-out (~7 MB total; 256B-aligned chunks)
    char* ws = (char*)d_ws;
    size_t off = 0;
    auto carve = [&](size_t bytes) -> void* {
        void* p = ws + off;
        off += (bytes + 255) & ~(size_t)255;
        return p;
    };
    float* pD1  = (float*)carve((size_t)Np * nSl1 * 3 * sizeof(float));
    int*   pJ1  = (int*)  carve((size_t)Np * nSl1 * 3 * sizeof(int));
    float* pD2  = (float*)carve((size_t)Nm * nSl2 * 3 * sizeof(float));
    int*   pJ2  = (int*)  carve((size_t)Nm * nSl2 * 3 * sizeof(int));
    int*   idx1 = (int*)  carve((size_t)Np * 3 * sizeof(int));
    float* w1   = (float*)carve((size_t)Np * 3 * sizeof(float));
    int*   idx2 = (int*)  carve((size_t)Nm * 3 * sizeof(int));
    float* w2   = (float*)carve((size_t)Nm * 3 * sizeof(float));
    float* piv  = (float*)carve((size_t)B * Np * C * sizeof(float));
    (void)ws_size; (void)n_in; (void)out_size;

    // Pass 1: pivotal queries vs mesh candidates (sliced) + merge
    {
        const int waves  = ((Np + 15) / 16) * nSl1;
        knn3_wmma_partial_kernel<<<(waves + 7) / 8, 256, 0, stream>>>(
            posP, Np, posM, Nm, nSl1, tps1, pD1, pJ1);
        knn3_merge_kernel<<<(Np + 255) / 256, 256, 0, stream>>>(
            posP, Np, posM, pD1, pJ1, nSl1, idx1, w1);
    }
    // Pass 2: mesh queries vs pivotal candidates + merge
    {
        const int waves  = ((Nm + 15) / 16) * nSl2;
        knn3_wmma_partial_kernel<<<(waves + 7) / 8, 256, 0, stream>>>(
            posM, Nm, posP, Np, nSl2, tps2, pD2, pJ2);
        knn3_merge_kernel<<<(Nm + 255) / 256, 256, 0, stream>>>(
            posM, Nm, posP, pD2, pJ2, nSl2, idx2, w2);
    }
    // Stage 2: fused LayerNorm + gather into piv
    {
        const int waves = B * Np;
        piv_gather_ln_kernel<<<(waves + 7) / 8, 256, 0, stream>>>(
            nf, gamma, beta, idx1, w1, piv, B, Np, Nm, C);
    }
    // Stage 3: gather piv back to mesh -> output
    {
        const int waves = B * Nm;
        mesh_gather_kernel<<<(waves + 7) / 8, 256, 0, stream>>>(
            piv, idx2, w2, (float*)d_out, B, Np, Nm, C);
    }
}